// model_75720273429211
// MI455X (gfx1250) — compile-verified
//
#include <hip/hip_runtime.h>

// FM scoring: gather 26x16 embeddings per row, linear dot + FM interaction via
// two f32 WMMA chains (xv = emb@v, sos = emb^2@v^2), sigmoid at the end.
// M=16 batch rows per wave, N=16 factors, K=416 reduced 4 at a time with
// V_WMMA_F32_16X16X4_F32 (exact f32, matches reference numerics).

typedef __attribute__((ext_vector_type(2))) float v2f;
typedef __attribute__((ext_vector_type(8))) float v8f;

#define N_FIELDS 26
#define VOCAB    100000
#define EMBED    16
#define NDIM     (N_FIELDS * EMBED)   // 416
#define KFAC     16
#define WAVES    8
#define VSTRIDE  418                  // padded row stride (floats) -> 8B-aligned ds_load_b64

__global__ __launch_bounds__(WAVES * 32) void fm_wmma_kernel(
    const int* __restrict__ X,        // [B, 26]
    const float* __restrict__ tables, // [26, 100000, 16]
    const float* __restrict__ W,      // [416]
    const float* __restrict__ bptr,   // [1]
    const float* __restrict__ v,      // [416, 16]
    float* __restrict__ out,          // [B]
    int ntiles)
{
    // v transposed: vshT[k][n] at k*VSTRIDE + n, so B-operand loads are contiguous in n.
    __shared__ float vshT[KFAC * VSTRIDE];   // ~26.8 KB
    __shared__ float wsh[NDIM];              // 1.7 KB

    const int tid = threadIdx.x;
    for (int i = tid; i < NDIM * KFAC; i += WAVES * 32) {
        int n = i >> 4, k = i & 15;
        vshT[k * VSTRIDE + n] = v[i];        // v is [n][k] row-major, i = n*16+k
    }
    for (int i = tid; i < NDIM; i += WAVES * 32) wsh[i] = W[i];
    __syncthreads();

    const int lane = tid & 31;
    const int wv   = tid >> 5;
    const int tile = blockIdx.x * WAVES + wv;
    if (tile >= ntiles) return;              // after barrier: safe

    const int half = lane >> 4;              // which K-pair this lane supplies
    const int mrow = lane & 15;              // A: M row   /  B: N column
    const int r0   = tile * 16;
    const int row  = r0 + mrow;

    const int* Xrow = X + row * N_FIELDS;
    const float* vrow = &vshT[mrow * VSTRIDE];

    v8f acc1 = {};  // xv  = emb   @ v
    v8f acc2 = {};  // sos = emb^2 @ v^2
    float linp = 0.0f;

    for (int f = 0; f < N_FIELDS; ++f) {
        const int idx = Xrow[f];
        const float* ep = tables + ((size_t)f * VOCAB + (size_t)idx) * EMBED;
        const int nf = f * EMBED;
#pragma unroll
        for (int c = 0; c < 4; ++c) {
            const int e  = 4 * c + 2 * half;     // K-lane slot within this chunk
            const int nx = nf + e;               // global reduction index
            v2f a  = *(const v2f*)(ep + e);                 // global_load_b64 (gather)
            v2f bv = *(const v2f*)(vrow + nx);              // ds_load_b64
            v2f wl = *(const v2f*)(&wsh[nx]);               // ds_load_b64
            v2f a2 = a * a;
            v2f b2 = bv * bv;
            acc1 = __builtin_amdgcn_wmma_f32_16x16x4_f32(
                false, a,  false, bv, (short)0, acc1, false, false);
            acc2 = __builtin_amdgcn_wmma_f32_16x16x4_f32(
                false, a2, false, b2, (short)0, acc2, false, false);
            linp += a.x * wl.x + a.y * wl.y;     // linear term rides along
        }
    }

    // interaction terms per (row, k): D layout = VGPR r, half h -> row r+8h, col = mrow
    float s[8];
#pragma unroll
    for (int r = 0; r < 8; ++r) s[r] = acc1[r] * acc1[r] - acc2[r];
    // reduce over the 16 factor columns (stays inside each 16-lane half-group)
#pragma unroll
    for (int r = 0; r < 8; ++r) {
        s[r] += __shfl_xor(s[r], 1, 32);
        s[r] += __shfl_xor(s[r], 2, 32);
        s[r] += __shfl_xor(s[r], 4, 32);
        s[r] += __shfl_xor(s[r], 8, 32);
    }

    // linear: lane l holds half the dot for row (l&15); combine halves,
    // then route row m = (l&7)+8*half to its writer lane.
    float linfull = linp + __shfl_xor(linp, 16, 32);
    const int mtar = (lane & 7) + 8 * half;
    const float lin_m = __shfl(linfull, mtar, 32);
    const float bias = bptr[0];

#pragma unroll
    for (int r = 0; r < 8; ++r) {
        if ((lane & 15) == r) {                  // lanes r and r+16 -> rows r, r+8
            float z = lin_m + bias + 0.5f * s[r];
            out[r0 + r + 8 * half] = 1.0f / (1.0f + __expf(-z));
        }
    }
}

extern "C" void kernel_launch(void* const* d_in, const int* in_sizes, int n_in,
                              void* d_out, int out_size, void* d_ws, size_t ws_size,
                              hipStream_t stream) {
    const int*   X      = (const int*)d_in[0];
    const float* tables = (const float*)d_in[1];
    const float* W      = (const float*)d_in[2];
    const float* b      = (const float*)d_in[3];
    const float* v      = (const float*)d_in[4];
    float*       out    = (float*)d_out;

    const int batch  = in_sizes[0] / N_FIELDS;   // 16384
    const int ntiles = batch / 16;               // 1024
    dim3 block(WAVES * 32);
    dim3 grid((ntiles + WAVES - 1) / WAVES);     // 128 blocks
    fm_wmma_kernel<<<grid, block, 0, stream>>>(X, tables, W, b, v, out, ntiles);
}